// SRNN_ALIF_fix_84009560310444
// MI455X (gfx1250) — compile-verified
//
#include <hip/hip_runtime.h>

typedef __attribute__((ext_vector_type(2))) float v2f;
typedef __attribute__((ext_vector_type(8))) float v8f;

#define B_J0_C  0.01f
#define BETA_C  1.8f
#define ALPHA_C 0.951229424500714f   // exp(-1/20)
#define RO_C    0.998334721126256f   // exp(-1/600)

#define HH 512
#define TT 500
#define NW 4   // waves per block, split over K

// Fused ALIF layer step:  C = A1 @ W1^T + A2 @ W2^T  (+bias1+bias2), then ALIF update.
// A1: [64 x K1] rows strided by a1_stride (offset a1_off), W1: [512 x K1] row-major.
// A2: s_prev [64 x 512], W2: [512 x 512]. One 16x16 output tile per block, 4 waves split K.
__global__ __launch_bounds__(128) void alif_layer_step(
    const float* __restrict__ A1, long a1_off, int a1_stride, int K1,
    const float* __restrict__ W1, const float* __restrict__ bias1,
    const float* __restrict__ A2,
    const float* __restrict__ W2, const float* __restrict__ bias2,
    float* __restrict__ mstate, float* __restrict__ bstate,
    float* __restrict__ s_new,
    float* __restrict__ out_mem, float* __restrict__ out_spk,
    float* __restrict__ out_th, int t)
{
    const int lane = threadIdx.x & 31;
    // Force wave id into an SGPR so K-loop bounds are scalar: loops compile to
    // s_cmp/s_cbranch with full EXEC (required for WMMA) instead of exec-masked
    // per-lane loops.
    const int wave = __builtin_amdgcn_readfirstlane(threadIdx.x >> 5);
    const int nbase = blockIdx.x * 16;   // hidden dim tile
    const int mbase = blockIdx.y * 16;   // batch tile

    // A fragment (16x4 f32): lane 0-15 hold M=lane; VGPR0=K0|K2, VGPR1=K1|K3 by lane half.
    // B fragment (4x16 f32): row K striped across lanes within a VGPR, lane halves K/K+2.
    const int rowA = mbase + (lane & 15);
    const int rowB = nbase + (lane & 15);
    const int kh = 2 * (lane >> 4);      // 0 for lanes 0-15, 2 for lanes 16-31

    v8f c = {};

    // GEMM 1: feed-forward input (K1 = 700 or 512)
    {
        const float* A1p = A1 + a1_off + (long)rowA * a1_stride + kh;
        const float* W1p = W1 + (long)rowB * K1 + kh;
        #pragma unroll 4
        for (int k = wave * 4; k < K1; k += 4 * NW) {
            v2f a = *(const v2f*)(A1p + k);
            v2f b = *(const v2f*)(W1p + k);
            c = __builtin_amdgcn_wmma_f32_16x16x4_f32(false, a, false, b,
                                                      (short)0, c, false, false);
        }
    }
    // GEMM 2: recurrent spikes (K2 = H = 512 -> exactly 32 iterations per wave)
    {
        const float* A2p = A2 + (long)rowA * HH + kh;
        const float* W2p = W2 + (long)rowB * HH + kh;
        #pragma unroll 4
        for (int k = wave * 4; k < HH; k += 4 * NW) {
            v2f a = *(const v2f*)(A2p + k);
            v2f b = *(const v2f*)(W2p + k);
            c = __builtin_amdgcn_wmma_f32_16x16x4_f32(false, a, false, b,
                                                      (short)0, c, false, false);
        }
    }

    // Reduce the 4 partial accumulators through LDS; wave 0 owns the epilogue.
    __shared__ float red[NW - 1][256];
    if (wave > 0) {
        #pragma unroll
        for (int v = 0; v < 8; ++v) red[wave - 1][v * 32 + lane] = c[v];
    }
    __syncthreads();

    if (wave == 0) {
        // C layout: VGPR v, lanes 0-15 -> M=v, lanes 16-31 -> M=v+8; N = lane&15
        const int nn = nbase + (lane & 15);
        const int mm0 = mbase + 8 * (lane >> 4);
        const float bsum = bias1[nn] + bias2[nn];
        const long idx0 = (long)mm0 * HH + nn;
        const long oidx0 = ((long)mm0 * TT + t) * HH + nn;
        #pragma unroll
        for (int v = 0; v < 8; ++v) {
            float acc = c[v] + red[0][v * 32 + lane]
                             + red[1][v * 32 + lane]
                             + red[2][v * 32 + lane];
            const long idx = idx0 + (long)v * HH;

            float h   = acc + bsum;
            float spk = A2[idx];                                  // s(t-1)
            float bb  = RO_C * bstate[idx] + (1.0f - RO_C) * spk; // adaptive var
            float Bth = B_J0_C + BETA_C * bb;
            float mem = mstate[idx] * ALPHA_C + (1.0f - ALPHA_C) * h - Bth * spk;
            float s   = (mem - Bth) > 0.0f ? 1.0f : 0.0f;

            mstate[idx] = mem;
            bstate[idx] = bb;
            s_new[idx]  = s;
            if (out_mem) {   // layer 3 also streams hidden outputs [B,T,H]
                const long oidx = oidx0 + (long)v * TT * HH;
                out_mem[oidx] = mem;
                out_spk[oidx] = s;
                out_th[oidx]  = Bth;
            }
        }
    }
}

// Readout: out[b,t,o] = bo[o] + dot(m3[b,:], Wo[o,:])   (64x20 outputs, K=512)
__global__ __launch_bounds__(256) void out_proj(
    const float* __restrict__ m3, const float* __restrict__ Wo,
    const float* __restrict__ bo, float* __restrict__ out, int t)
{
    const int OUTN = 20;
    int i = blockIdx.x * blockDim.x + threadIdx.x;
    if (i >= 64 * OUTN) return;
    int b = i / OUTN, o = i % OUTN;
    const float* mrow = m3 + (long)b * HH;
    const float* wrow = Wo + (long)o * HH;
    float acc = 0.0f;
    #pragma unroll 4
    for (int k = 0; k < HH; k += 4) {
        float4 mv = *(const float4*)(mrow + k);
        float4 wv = *(const float4*)(wrow + k);
        acc += mv.x * wv.x + mv.y * wv.y + mv.z * wv.z + mv.w * wv.w;
    }
    out[((long)b * TT + t) * OUTN + o] = bo[o] + acc;
}

// Per-layer workspace layout: [m | b | s_ping | s_pong], each 64*512 floats.
__global__ __launch_bounds__(256) void init_state(float* __restrict__ ws)
{
    const long N = 64L * HH;
    long i = (long)blockIdx.x * blockDim.x + threadIdx.x;
    if (i >= N) return;
    #pragma unroll
    for (int l = 0; l < 3; ++l) {
        float* base = ws + (long)l * 4 * N;
        base[0 * N + i] = 0.0f;     // m
        base[1 * N + i] = B_J0_C;   // b
        base[2 * N + i] = 0.0f;     // s ping
        base[3 * N + i] = 0.0f;     // s pong
    }
}

extern "C" void kernel_launch(void* const* d_in, const int* in_sizes, int n_in,
                              void* d_out, int out_size, void* d_ws, size_t ws_size,
                              hipStream_t stream)
{
    const float* x   = (const float*)d_in[0];
    const float* Wi1 = (const float*)d_in[1];
    const float* bi1 = (const float*)d_in[2];
    const float* Wh1 = (const float*)d_in[3];
    const float* bh1 = (const float*)d_in[4];
    const float* Wi2 = (const float*)d_in[5];
    const float* bi2 = (const float*)d_in[6];
    const float* Wh2 = (const float*)d_in[7];
    const float* bh2 = (const float*)d_in[8];
    const float* Wi3 = (const float*)d_in[9];
    const float* bi3 = (const float*)d_in[10];
    const float* Wh3 = (const float*)d_in[11];
    const float* bh3 = (const float*)d_in[12];
    const float* Wo  = (const float*)d_in[13];
    const float* bo  = (const float*)d_in[14];

    const int Bn = 64, T = TT, IN = 700, H = HH, OUTN = 20;
    const long N = (long)Bn * H;

    float* ws = (float*)d_ws;
    float* m1 = ws + 0L * 4 * N; float* b1 = m1 + N; float* s1a = m1 + 2 * N; float* s1b = m1 + 3 * N;
    float* m2 = ws + 1L * 4 * N; float* b2 = m2 + N; float* s2a = m2 + 2 * N; float* s2b = m2 + 3 * N;
    float* m3 = ws + 2L * 4 * N; float* b3 = m3 + N; float* s3a = m3 + 2 * N; float* s3b = m3 + 3 * N;

    float* out   = (float*)d_out;                      // [B,T,OUT]
    float* o_spk = out   + (long)Bn * T * OUTN;        // [B,T,H]
    float* o_mem = o_spk + (long)Bn * T * H;           // [B,T,H]
    float* o_th  = o_mem + (long)Bn * T * H;           // [B,T,H]

    init_state<<<(int)((N + 255) / 256), 256, 0, stream>>>(ws);

    dim3 grid(H / 16, Bn / 16);   // 32 x 4 tiles
    for (int t = 0; t < T; ++t) {
        float* s1p = (t & 1) ? s1b : s1a;  float* s1n = (t & 1) ? s1a : s1b;
        float* s2p = (t & 1) ? s2b : s2a;  float* s2n = (t & 1) ? s2a : s2b;
        float* s3p = (t & 1) ? s3b : s3a;  float* s3n = (t & 1) ? s3a : s3b;

        // Layer 1: x_t @ Wi1^T + s1(t-1) @ Wh1^T
        alif_layer_step<<<grid, 128, 0, stream>>>(
            x, (long)t * IN, T * IN, IN, Wi1, bi1,
            s1p, Wh1, bh1, m1, b1, s1n,
            nullptr, nullptr, nullptr, t);
        // Layer 2: s1(t) @ Wi2^T + s2(t-1) @ Wh2^T
        alif_layer_step<<<grid, 128, 0, stream>>>(
            s1n, 0L, H, H, Wi2, bi2,
            s2p, Wh2, bh2, m2, b2, s2n,
            nullptr, nullptr, nullptr, t);
        // Layer 3: s2(t) @ Wi3^T + s3(t-1) @ Wh3^T (+ hidden outputs)
        alif_layer_step<<<grid, 128, 0, stream>>>(
            s2n, 0L, H, H, Wi3, bi3,
            s3p, Wh3, bh3, m3, b3, s3n,
            o_mem, o_spk, o_th, t);
        // Readout
        out_proj<<<(Bn * OUTN + 255) / 256, 256, 0, stream>>>(m3, Wo, bo, out, t);
    }
}